// Softmax_73521250173287
// MI455X (gfx1250) — compile-verified
//
#include <hip/hip_runtime.h>
#include <hip/hip_bf16.h>
#include <cstdint>
#include <math.h>

// Problem constants (from the reference file).
#define N_TOTAL 67108864               // 2^26 elements
#define B_SEG   16384                  // number of segments
#define N4      (N_TOTAL / 4)          // 16,777,216 float4 tiles

// Launch geometry: every thread runs EXACTLY ITERS iterations (no remainder)
// -> uniform EXEC for the wave32 scans and a clean async software pipeline.
#define MAIN_BLOCKS 4096
#define THREADS     256
#define STRIDE      (MAIN_BLOCKS * THREADS)   // 1,048,576
#define ITERS       (N4 / STRIDE)             // 16
static_assert(N4 % STRIDE == 0, "uniform trip count required");

// ---------------------------------------------------------------------------
// CDNA5 async global->LDS copy (tracked by ASYNCcnt, wave-local completion).
__device__ __forceinline__ void async_load_b128(uint32_t lds_addr, const void* gptr) {
    asm volatile("global_load_async_to_lds_b128 %0, %1, off"
                 :
                 : "v"(lds_addr), "v"((uint64_t)(uintptr_t)gptr)
                 : "memory");
}
__device__ __forceinline__ void wait_async_le1() {
    asm volatile("s_wait_asynccnt 0x1" ::: "memory");
}
__device__ __forceinline__ void wait_async_le0() {
    asm volatile("s_wait_asynccnt 0x0" ::: "memory");
}

// ---------------------------------------------------------------------------
// Online-softmax merge: (m, a) <- (m, a) ⊕ (om, oa), where a is the running
// sum of exp(x - m). Since the new max is max(m, om), exactly one rescale
// factor is exp(0)==1, so only ONE exponential is needed; the rescale factor
// uses the hardware v_exp_f32 (__expf, rel err ~5e-7 — it only scales partial
// sums). Merging into the (-inf, 0) init cell is clean: 0*__expf(-inf) == 0.
__device__ __forceinline__ void online_merge(float& m, float& a, float om, float oa) {
    if (om > m) {
        a = a * __expf(m - om) + oa;
        m = om;
    } else {
        a = a + oa * __expf(om - m);
    }
}

// Packed cell: low word = running max (f32 bits), high = running sum (f32 bits).
__device__ __forceinline__ unsigned long long pack_ms(float m, float s) {
    return ((unsigned long long)__float_as_uint(s) << 32) |
           (unsigned long long)__float_as_uint(m);
}

// Atomic online merge of a partial (m_i, a_i) into the per-segment cell via a
// 64-bit CAS loop (GLOBAL_ATOMIC_CMPSWAP_B64). ~1 merge per 128 elements.
__device__ __forceinline__ void atomicMergeMS(unsigned long long* addr,
                                              float m_i, float a_i) {
    unsigned long long cur = *(volatile unsigned long long*)addr;
    while (true) {
        float m = __uint_as_float((unsigned)(cur & 0xffffffffu));
        float a = __uint_as_float((unsigned)(cur >> 32));
        online_merge(m, a, m_i, a_i);
        unsigned long long desired = pack_ms(m, a);
        unsigned long long seen = atomicCAS(addr, cur, desired);
        if (seen == cur) return;
        cur = seen;
    }
}

// ---------------------------------------------------------------------------
__global__ __launch_bounds__(THREADS) void k_init(unsigned long long* __restrict__ tab) {
    int i = blockIdx.x * blockDim.x + threadIdx.x;
    if (i < B_SEG) {
        tab[i] = pack_ms(-INFINITY, 0.0f);
    }
}

// ---------------------------------------------------------------------------
// Fused pass: per-segment (max, sumexp) in ONE sweep of x/seg.
// x streamed via double-buffered async-to-LDS. Sorted ids -> in-thread online
// run compression, then a wave32 segmented inclusive scan whose combine op is
// the associative online-softmax merge; only run-tail lanes hit the CAS cell.
__global__ __launch_bounds__(THREADS) void k_online(const float4* __restrict__ x4,
                                                    const int4* __restrict__ s4,
                                                    unsigned long long* __restrict__ tab) {
    __shared__ float4 tile[2][THREADS];   // 8 KB double buffer, 16 B/lane
    const int lane = threadIdx.x & 31;
    const uint32_t lds0 = (uint32_t)(uintptr_t)&tile[0][threadIdx.x];
    const uint32_t lds1 = (uint32_t)(uintptr_t)&tile[1][threadIdx.x];

    int i = blockIdx.x * blockDim.x + threadIdx.x;
    async_load_b128(lds0, x4 + i);        // prologue: tile for iteration 0

    #pragma unroll 1
    for (int k = 0; k < ITERS; ++k) {
        if (k + 1 < ITERS) {
            async_load_b128(((k + 1) & 1) ? lds1 : lds0, x4 + i + STRIDE);
        }
        int4 s = s4[i];                   // overlaps the async transfers
        if (k + 1 < ITERS) wait_async_le1(); else wait_async_le0();
        float4 v = tile[k & 1][threadIdx.x];

        // In-thread online accumulation with run compression. Element
        // contributions use the precise expf; rescales use __expf.
        int   seg = s.x;
        float m   = v.x;
        float a   = 1.0f;                 // sum of exp(elem - m) for this run
        if (s.y == seg) {
            if (v.y > m) { a = a * __expf(m - v.y) + 1.0f; m = v.y; }
            else         { a += expf(v.y - m); }
        } else { atomicMergeMS(&tab[seg], m, a); seg = s.y; m = v.y; a = 1.0f; }
        if (s.z == seg) {
            if (v.z > m) { a = a * __expf(m - v.z) + 1.0f; m = v.z; }
            else         { a += expf(v.z - m); }
        } else { atomicMergeMS(&tab[seg], m, a); seg = s.z; m = v.z; a = 1.0f; }
        if (s.w == seg) {
            if (v.w > m) { a = a * __expf(m - v.w) + 1.0f; m = v.w; }
            else         { a += expf(v.w - m); }
        } else { atomicMergeMS(&tab[seg], m, a); seg = s.w; m = v.w; a = 1.0f; }

        // Wave32 segmented inclusive scan with the online-merge combine
        // (valid: sorted ids make equal-id lanes contiguous; op associative).
        #pragma unroll
        for (int off = 1; off < 32; off <<= 1) {
            int   oseg = __shfl_up(seg, off, 32);
            float om   = __shfl_up(m, off, 32);
            float oa   = __shfl_up(a, off, 32);
            if (lane >= off && oseg == seg) {
                online_merge(m, a, om, oa);
            }
        }
        int nseg = __shfl_down(seg, 1, 32);
        if (lane == 31 || nseg != seg) {
            atomicMergeMS(&tab[seg], m, a);
        }
        i += STRIDE;
    }
}

// ---------------------------------------------------------------------------
// Tiny pass: (m, s) -> (m, 1/s) in place, so the final sweep gathers a single
// float2 per element and is multiply-only. Empty segments -> inf, never read.
__global__ __launch_bounds__(THREADS) void k_recip(float2* __restrict__ tab) {
    int i = blockIdx.x * blockDim.x + threadIdx.x;
    if (i < B_SEG) {
        float2 t = tab[i];
        t.y = 1.0f / t.y;
        tab[i] = t;
    }
}

// ---------------------------------------------------------------------------
// Final pass: out = exp(x - m[seg]) * inv[seg]; x streamed via the same
// double-buffered async-to-LDS pipeline; (m, inv) gathered as one 8 B load.
__global__ __launch_bounds__(THREADS) void k_scale(const float4* __restrict__ x4,
                                                   const int4* __restrict__ s4,
                                                   const float2* __restrict__ tab,
                                                   float4* __restrict__ out4) {
    __shared__ float4 tile[2][THREADS];
    const uint32_t lds0 = (uint32_t)(uintptr_t)&tile[0][threadIdx.x];
    const uint32_t lds1 = (uint32_t)(uintptr_t)&tile[1][threadIdx.x];

    int i = blockIdx.x * blockDim.x + threadIdx.x;
    async_load_b128(lds0, x4 + i);

    #pragma unroll 1
    for (int k = 0; k < ITERS; ++k) {
        if (k + 1 < ITERS) {
            async_load_b128(((k + 1) & 1) ? lds1 : lds0, x4 + i + STRIDE);
        }
        int4 s = s4[i];
        if (k + 1 < ITERS) wait_async_le1(); else wait_async_le0();
        float4 v = tile[k & 1][threadIdx.x];

        float2 tx = tab[s.x];
        float2 ty = tab[s.y];
        float2 tz = tab[s.z];
        float2 tw = tab[s.w];

        float4 r;
        r.x = expf(v.x - tx.x) * tx.y;
        r.y = expf(v.y - ty.x) * ty.y;
        r.z = expf(v.z - tz.x) * tz.y;
        r.w = expf(v.w - tw.x) * tw.y;
        out4[i] = r;
        i += STRIDE;
    }
}

// ---------------------------------------------------------------------------
extern "C" void kernel_launch(void* const* d_in, const int* in_sizes, int n_in,
                              void* d_out, int out_size, void* d_ws, size_t ws_size,
                              hipStream_t stream) {
    (void)in_sizes; (void)n_in; (void)out_size; (void)ws_size;

    const float* x   = (const float*)d_in[0];
    const int*   seg = (const int*)d_in[1];
    float*       out = (float*)d_out;

    unsigned long long* tab = (unsigned long long*)d_ws;   // B_SEG x (m, s) cells = 128 KB

    const float4* x4 = (const float4*)x;
    const int4*   s4 = (const int4*)seg;

    const int small_blocks = B_SEG / THREADS;   // 64

    k_init  <<<small_blocks, THREADS, 0, stream>>>(tab);
    k_online<<<MAIN_BLOCKS, THREADS, 0, stream>>>(x4, s4, tab);
    k_recip <<<small_blocks, THREADS, 0, stream>>>((float2*)tab);
    k_scale <<<MAIN_BLOCKS, THREADS, 0, stream>>>(x4, s4, (const float2*)tab, (float4*)out);
}